// KerasLayer_58695023067329
// MI455X (gfx1250) — compile-verified
//
#include <hip/hip_runtime.h>

#ifndef __has_builtin
#define __has_builtin(x) 0
#endif

#define OSC_N 256
#define T_N   2048
#define DT    0.025f
#define DT6   (0.025f / 6.0f)
#define TWO_PI 6.283185307179586f

// zdot(z, p) with the reference's constants folded:
//   zdot = z*(i*w - f*|z|^2) + (1+i)*p*f * 1/((1-zr) + i*zi)
// Chain kept short: 2 FMAs for |z|^2, FMA-form complex mul, one v_rcp_f32.
__device__ __forceinline__ void zdot(float zr, float zi, float p, float w, float f,
                                     float& dr, float& di) {
    float m2  = __builtin_fmaf(zr, zr, zi * zi);
    float g   = -f * m2;                          // real part of (A + B*|z|^2)
    float t1r = __builtin_fmaf(zr, g, -(zi * w)); // z * (g + i*w)
    float t1i = __builtin_fmaf(zr, w,  zi * g);
    float ur  = 1.0f - zr;                        // 1 - conj(z) = ur + i*zi
    float den = __builtin_fmaf(ur, ur, zi * zi);
    float inv = __builtin_amdgcn_rcpf(den);       // v_rcp_f32, off IEEE-div path
    float ar  =  ur * inv;
    float ai  = -zi * inv;
    float pf  = p * f;                            // (1+i)*(ar+i*ai) = (ar-ai)+i(ar+ai)
    dr = __builtin_fmaf(pf, ar - ai, t1r);
    di = __builtin_fmaf(pf, ar + ai, t1i);
}

__global__ __launch_bounds__(OSC_N) void grfnn_rk4_kernel(const float* __restrict__ x,
                                                          float* __restrict__ out) {
    __shared__ float p[T_N];                      // 8 KiB: per-batch drive terms
    const int b   = blockIdx.x;
    const int osc = threadIdx.x;
    const float* xb = x + (size_t)b * T_N;

#if defined(__gfx1250__) && __has_builtin(__builtin_amdgcn_global_load_async_to_lds_b32)
    // CDNA5 async global->LDS staging of this batch's input series.
    #pragma unroll
    for (int i = 0; i < T_N / OSC_N; ++i) {
        const int idx = osc + i * OSC_N;
        __builtin_amdgcn_global_load_async_to_lds_b32(
            (__attribute__((address_space(1))) int*)(xb + idx),
            (__attribute__((address_space(3))) int*)(p + idx),
            0, 0);
    }
    #if __has_builtin(__builtin_amdgcn_s_wait_asynccnt)
    __builtin_amdgcn_s_wait_asynccnt(0);
    #else
    asm volatile("s_wait_asynccnt 0x0" ::: "memory");
    #endif
    // Each lane converts exactly the entries it loaded: x -> p = x/(1-x).
    #pragma unroll
    for (int i = 0; i < T_N / OSC_N; ++i) {
        const int idx = osc + i * OSC_N;
        float xv = p[idx];
        p[idx] = xv * __builtin_amdgcn_rcpf(1.0f - xv);
    }
#else
    for (int idx = osc; idx < T_N; idx += OSC_N) {
        float xv = xb[idx];
        p[idx] = xv * __builtin_amdgcn_rcpf(1.0f - xv);
    }
#endif
    __syncthreads();

    // f = logspace(0.125, 8, 256)[osc] = 2^(-3 + 6*osc/255)
    const float f = __builtin_exp2f(__builtin_fmaf(6.0f / 255.0f, (float)osc, -3.0f));
    const float w = TWO_PI * f;

    float zr = 0.0f, zi = 0.0f;
    float* orow = out + ((size_t)(b * OSC_N + osc)) * (size_t)T_N;
    orow[0] = 0.0f;                               // |z0| = 0

    float p_i = p[0];
    for (int t = 0; t < T_N - 1; ++t) {
        const float p_n = p[t + 1];               // LDS broadcast (all lanes same addr)
        float k1r, k1i, k2r, k2i, k3r, k3i, k4r, k4i;
        zdot(zr, zi, p_i, w, f, k1r, k1i);
        zdot(__builtin_fmaf(0.5f * DT, k1r, zr), __builtin_fmaf(0.5f * DT, k1i, zi),
             p_i, w, f, k2r, k2i);
        zdot(__builtin_fmaf(0.5f * DT, k2r, zr), __builtin_fmaf(0.5f * DT, k2i, zi),
             p_i, w, f, k3r, k3i);
        zdot(__builtin_fmaf(DT, k3r, zr), __builtin_fmaf(DT, k3i, zi),
             p_n, w, f, k4r, k4i);
        zr = __builtin_fmaf(DT6, k1r + 2.0f * (k2r + k3r) + k4r, zr);
        zi = __builtin_fmaf(DT6, k1i + 2.0f * (k2i + k3i) + k4i, zi);
        orow[t + 1] = __builtin_amdgcn_sqrtf(__builtin_fmaf(zr, zr, zi * zi));
        p_i = p_n;
    }
}

extern "C" void kernel_launch(void* const* d_in, const int* in_sizes, int n_in,
                              void* d_out, int out_size, void* d_ws, size_t ws_size,
                              hipStream_t stream) {
    const float* x  = (const float*)d_in[0];
    float* out      = (float*)d_out;
    const int B     = in_sizes[0] / T_N;          // 32 for the reference shapes
    hipLaunchKernelGGL(grfnn_rk4_kernel, dim3(B), dim3(OSC_N), 0, stream, x, out);
}